// FusionNetwork_42958262895310
// MI455X (gfx1250) — compile-verified
//
#include <hip/hip_runtime.h>
#include <cstdint>

typedef float v4f __attribute__((ext_vector_type(4)));

// Problem constants (match setup_inputs): B=16, L=2048, D=1024, H=512
#define BB 16
#define LL 2048
#define DD 1024
#define HH 512
#define LCHUNKS 16          // pass-1 split of L
#define ROWS_PER_CHUNK (LL / LCHUNKS)   // 128
#define ROWS_PER_BLK 8      // pass-3 rows per block

// ---------------- Pass 1: partial column sums over L chunks ----------------
// grid = BB*LCHUNKS blocks, 256 threads; thread t owns columns [4t, 4t+4).
__global__ void __launch_bounds__(256)
colsum_partial_kernel(const float* __restrict__ inp, float* __restrict__ partial) {
    const int bx    = blockIdx.x;          // 0..255
    const int b     = bx >> 4;
    const int chunk = bx & 15;
    const int t     = threadIdx.x;         // 0..255

    const float* p = inp + ((size_t)b * LL + (size_t)chunk * ROWS_PER_CHUNK) * DD + 4 * t;
    v4f acc = {0.0f, 0.0f, 0.0f, 0.0f};
#pragma unroll 8
    for (int i = 0; i < ROWS_PER_CHUNK; ++i) {
        v4f x = *(const v4f*)(p + (size_t)i * DD);   // RT load: warms the 192MB L2
        acc += x;
    }
    *(v4f*)(partial + (size_t)bx * DD + 4 * t) = acc;
}

// ---------------- Pass 2: reduce partials -> column mean ----------------
// grid = 16 blocks, 256 threads; linear index over B*D/4 float4s.
__global__ void __launch_bounds__(256)
colmean_reduce_kernel(const float* __restrict__ partial, float* __restrict__ colmean) {
    const int idx = blockIdx.x * 256 + threadIdx.x;  // 0..4095
    const int b   = idx >> 8;                        // 0..15
    const int d4  = (idx & 255) * 4;                 // 0..1020

    v4f acc = {0.0f, 0.0f, 0.0f, 0.0f};
#pragma unroll
    for (int c = 0; c < LCHUNKS; ++c)
        acc += *(const v4f*)(partial + ((size_t)(b * LCHUNKS + c)) * DD + d4);
    acc *= (1.0f / (float)LL);
    *(v4f*)(colmean + (size_t)b * DD + d4) = acc;
}

// ---------------- Pass 3: emit [inp | colmean | concat(h0,h1)] ----------------
// grid = BB*(LL/ROWS_PER_BLK) = 4096 blocks, 256 threads.
// LDS staging of the two broadcast rows via CDNA5 async global->LDS DMA,
// then 8 output rows of NT (cache-rinsing) 128-bit stores per block.
__global__ void __launch_bounds__(256)
emit_out_kernel(const float* __restrict__ inp,
                const float* __restrict__ colmean,
                const float* __restrict__ h0,
                const float* __restrict__ h1,
                float* __restrict__ out) {
    __shared__ v4f smem[512];                // [0:256) colmean row, [256:512) hcat row

    const int bx = blockIdx.x;
    const int b  = bx >> 8;                  // 0..15
    const int l0 = (bx & 255) * ROWS_PER_BLK;
    const int t  = threadIdx.x;              // 0..255

    // LDS byte offsets: low 32 bits of the flat LDS address are the LDS offset
    // (ISA 10.2).
    unsigned lds_cm = (unsigned)(size_t)&smem[t];
    unsigned lds_h  = (unsigned)(size_t)&smem[256 + t];

    const float* src_cm = colmean + (size_t)b * DD + 4 * t;
    const float* src_h  = (t < 128) ? (h0 + (size_t)b * HH + 4 * t)
                                    : (h1 + (size_t)b * HH + 4 * t - HH);

    // CDNA5 async tensor-path: global -> LDS, tracked by ASYNCcnt.
    asm volatile("global_load_async_to_lds_b128 %0, %1, off"
                 :: "v"(lds_cm), "v"((unsigned long long)(size_t)src_cm) : "memory");
    asm volatile("global_load_async_to_lds_b128 %0, %1, off"
                 :: "v"(lds_h), "v"((unsigned long long)(size_t)src_h) : "memory");
    asm volatile("s_wait_asynccnt 0" ::: "memory");
    __syncthreads();

    // Broadcast segments live in registers for all 8 rows.
    const v4f cm = smem[t];
    const v4f hh = smem[256 + t];

    const float* ip = inp + ((size_t)b * LL + l0) * DD + 4 * t;
    float*       op = out + ((size_t)b * LL + l0) * (2 * DD + 2 * HH) + 4 * t;

#pragma unroll
    for (int r = 0; r < ROWS_PER_BLK; ++r) {
        // Last-use read of inp (expected L2 hit from pass 1), NT so it rinses out.
        v4f a = __builtin_nontemporal_load((const v4f*)(ip + (size_t)r * DD));
        // 403MB output stream: NT stores keep inp resident in L2.
        __builtin_nontemporal_store(a,  (v4f*)(op + (size_t)r * 3072));
        __builtin_nontemporal_store(cm, (v4f*)(op + (size_t)r * 3072 + 1024));
        __builtin_nontemporal_store(hh, (v4f*)(op + (size_t)r * 3072 + 2048));
    }
}

extern "C" void kernel_launch(void* const* d_in, const int* in_sizes, int n_in,
                              void* d_out, int out_size, void* d_ws, size_t ws_size,
                              hipStream_t stream) {
    const float* hidden0 = (const float*)d_in[0];   // [16, 512]
    const float* hidden1 = (const float*)d_in[1];   // [16, 512]
    const float* inp     = (const float*)d_in[2];   // [16, 2048, 1024]
    float*       out     = (float*)d_out;           // [16, 2048, 3072]

    // Workspace layout: partial sums [256][1024] (1 MB), then colmean [16][1024] (64 KB).
    float* partial = (float*)d_ws;
    float* colmean = partial + (size_t)BB * LCHUNKS * DD;

    colsum_partial_kernel<<<BB * LCHUNKS, 256, 0, stream>>>(inp, partial);
    colmean_reduce_kernel<<<(BB * DD / 4) / 256, 256, 0, stream>>>(partial, colmean);
    emit_out_kernel<<<BB * (LL / ROWS_PER_BLK), 256, 0, stream>>>(inp, colmean, hidden0, hidden1, out);
}